// SGNS_25958782337670
// MI455X (gfx1250) — compile-verified
//
#include <hip/hip_runtime.h>
#include <hip/hip_bf16.h>
#include <stdint.h>

#define VOCAB   100000
#define DIM     128
#define BATCH   16384
#define NEG_K   10
#define WPB     4              // waves per block (wave32)
#define ROWSTRIDE 132          // 128 + 4 pad floats -> bank-conflict-free A reads
#define UTILE   (16 * ROWSTRIDE)
#define WAVE_LDS (UTILE + ROWSTRIDE)   // 16-row U tile + v_c row, per wave
#define SCAN_B  256

typedef __attribute__((ext_vector_type(2))) float v2f;
typedef __attribute__((ext_vector_type(8))) float v8f;

#if __has_builtin(__builtin_amdgcn_global_load_async_to_lds_b128)
#define USE_ASYNC_LDS 1
#else
#define USE_ASYNC_LDS 0
#endif

#if USE_ASYNC_LDS
typedef int v4i __attribute__((vector_size(16)));
typedef v4i __attribute__((address_space(1))) as1_v4i;   // global
typedef v4i __attribute__((address_space(3))) as3_v4i;   // LDS
#endif

// 16B global -> LDS copy, per lane. Async (GLOBAL_LOAD_ASYNC_TO_LDS_B128,
// ASYNCcnt-tracked, no VGPR round trip) when the builtin exists; else sync.
__device__ __forceinline__ void cp16_g2l(const float* __restrict__ g, float* l) {
#if USE_ASYNC_LDS
    as1_v4i* gp = (as1_v4i*)(uintptr_t)(const void*)g;
    as3_v4i* lp = (as3_v4i*)(uint32_t)(uintptr_t)(void*)l;   // low 32 bits = LDS offset
    __builtin_amdgcn_global_load_async_to_lds_b128(gp, lp, 0, 0);
#else
    *(float4*)l = *(const float4*)g;
#endif
}

__device__ __forceinline__ void wait_async_done() {
#if USE_ASYNC_LDS
#if __has_builtin(__builtin_amdgcn_s_wait_asynccnt)
    __builtin_amdgcn_s_wait_asynccnt(0);
#else
    asm volatile("s_wait_asynccnt 0x0" ::: "memory");
#endif
#endif
}

__device__ __forceinline__ uint64_t splitmix64(uint64_t x) {
    x += 0x9E3779B97F4A7C15ULL;
    x = (x ^ (x >> 30)) * 0xBF58476D1CE4E5B9ULL;
    x = (x ^ (x >> 27)) * 0x94D049BB133111EBULL;
    return x ^ (x >> 31);
}

__device__ __forceinline__ float logsig(float x) {
    // log(sigmoid(x)) = min(x,0) - log(1 + exp(-|x|)); hardware transcendentals
    return fminf(x, 0.0f) - __logf(1.0f + __expf(-fabsf(x)));
}

// ---------------- CDF build (unigram75 -> inclusive prefix sum) ----------------

__global__ void k_partial_sums(const float* __restrict__ p, float* __restrict__ bsum) {
    __shared__ float s[SCAN_B];
    int i = blockIdx.x * SCAN_B + threadIdx.x;
    s[threadIdx.x] = (i < VOCAB) ? p[i] : 0.0f;
    __syncthreads();
    for (int off = SCAN_B / 2; off > 0; off >>= 1) {
        if (threadIdx.x < off) s[threadIdx.x] += s[threadIdx.x + off];
        __syncthreads();
    }
    if (threadIdx.x == 0) bsum[blockIdx.x] = s[0];
}

__global__ void k_scan_bsums(float* __restrict__ bsum, int nb) {
    // tiny exclusive scan, single thread (nb = 391)
    float run = 0.0f;
    for (int i = 0; i < nb; ++i) { float t = bsum[i]; bsum[i] = run; run += t; }
}

__global__ void k_cdf(const float* __restrict__ p, const float* __restrict__ bsum,
                      float* __restrict__ cdf) {
    __shared__ float s[SCAN_B];
    int i = blockIdx.x * SCAN_B + threadIdx.x;
    s[threadIdx.x] = (i < VOCAB) ? p[i] : 0.0f;
    __syncthreads();
    // inclusive Hillis-Steele scan within the 256-chunk
    for (int off = 1; off < SCAN_B; off <<= 1) {
        float t = (threadIdx.x >= off) ? s[threadIdx.x - off] : 0.0f;
        __syncthreads();
        s[threadIdx.x] += t;
        __syncthreads();
    }
    if (i < VOCAB) cdf[i] = s[threadIdx.x] + bsum[blockIdx.x];
}

__global__ void k_zero_out(float* __restrict__ out) {
    if (threadIdx.x == 0) out[0] = 0.0f;
}

// ---------------- main SGNS kernel: one wave32 per batch element ----------------

__global__ void __launch_bounds__(WPB * 32)
k_sgns(const int* __restrict__ centers, const int* __restrict__ pos,
       const float* __restrict__ embed_in, const float* __restrict__ embed_out,
       const float* __restrict__ cdf, float* __restrict__ out) {
    __shared__ __align__(16) float smem[WPB * WAVE_LDS];
    __shared__ float wsum[WPB];

    const int lane = threadIdx.x & 31;
    const int wib  = threadIdx.x >> 5;
    const int b    = blockIdx.x * WPB + wib;

    float* Ubase = smem + wib * WAVE_LDS;   // 16 x ROWSTRIDE tile: [u_o; u_k*10; 0*5]
    float* vbase = Ubase + UTILE;           // v_c row (128 floats)

    // ---- pick 11 row indices: lane 0 = positive, lanes 1..10 = sampled negatives
    int rowidx = 0;
    if (lane == 0) {
        rowidx = pos[b];
    } else if (lane <= NEG_K) {
        const float total = cdf[VOCAB - 1];
        uint64_t h = splitmix64(((uint64_t)b << 8) ^ (uint64_t)lane ^ 0x5DEECE66DULL);
        float u = (float)(h >> 40) * (1.0f / 16777216.0f);   // [0,1)
        float target = u * total;
        int lo = 0, hi = VOCAB - 1;
        #pragma unroll 1
        while (lo < hi) {                                    // 17 iters, L2-resident CDF
            int mid = (lo + hi) >> 1;
            if (cdf[mid] < target) lo = mid + 1; else hi = mid;
        }
        rowidx = lo;
    }

    // ---- stage gathered rows straight into LDS (async-to-LDS on CDNA5),
    //      512B fully coalesced per row, one B128 per lane per row
    #pragma unroll 1
    for (int r = 0; r < 1 + NEG_K; ++r) {
        int idx = __shfl(rowidx, r, 32);
        cp16_g2l(embed_out + (size_t)idx * DIM + lane * 4,
                 Ubase + r * ROWSTRIDE + lane * 4);
    }
    int c = centers[b];
    cp16_g2l(embed_in + (size_t)c * DIM + lane * 4, vbase + lane * 4);

    float4 z4 = make_float4(0.f, 0.f, 0.f, 0.f);
    #pragma unroll
    for (int r = 1 + NEG_K; r < 16; ++r)
        *(float4*)(Ubase + r * ROWSTRIDE + lane * 4) = z4;   // zero-pad rows 11..15

    wait_async_done();      // ASYNCcnt -> 0: all async LDS writes landed
    __syncthreads();

    // ---- 16x16 score tile via V_WMMA_F32_16X16X4_F32, accumulate over D in K=4 steps
    // A (16x4 f32): lanes 0-15 hold M=row, K={k0,k0+1}; lanes 16-31 hold K={k0+2,k0+3}
    // B (4x16 f32): v_c broadcast across all 16 columns -> same-address LDS broadcast
    const int half = lane >> 4;
    const int row  = lane & 15;
    v8f acc = {};
    #pragma unroll 4
    for (int k0 = 0; k0 < DIM; k0 += 4) {
        v2f a = *(const v2f*)(Ubase + row * ROWSTRIDE + k0 + 2 * half);
        v2f bf;
        bf.x = vbase[k0 + half];
        bf.y = vbase[k0 + 2 + half];
        acc = __builtin_amdgcn_wmma_f32_16x16x4_f32(false, a, false, bf,
                                                    (short)0, acc, false, false);
    }

    // C layout: lanes 0-15 VGPR j = row M=j; lanes 16-31 VGPR j = row M=j+8.
    // All 16 columns identical -> read column via lane 0 / lane 16.
    float contrib;
    if (lane < 16) {
        contrib = logsig(acc[0]);                 // positive score (row 0)
        #pragma unroll
        for (int j = 1; j < 8; ++j) contrib += logsig(-acc[j]);   // negs 1..7
    } else {
        contrib = logsig(-acc[0]) + logsig(-acc[1]) + logsig(-acc[2]); // negs 8..10
    }
    float hi_part = __shfl(contrib, 16, 32);      // lane 16's sum, broadcast
    if (lane == 0) wsum[wib] = -(contrib + hi_part);

    __syncthreads();
    if (threadIdx.x == 0) {
        float s = 0.0f;
        #pragma unroll
        for (int w = 0; w < WPB; ++w) s += wsum[w];
        atomicAdd(out, s * (1.0f / (float)BATCH));
    }
}

// ---------------- launch ----------------

extern "C" void kernel_launch(void* const* d_in, const int* in_sizes, int n_in,
                              void* d_out, int out_size, void* d_ws, size_t ws_size,
                              hipStream_t stream) {
    const int*   centers   = (const int*)d_in[0];
    const int*   pos       = (const int*)d_in[1];
    const float* embed_in  = (const float*)d_in[2];
    const float* embed_out = (const float*)d_in[3];
    const float* unigram75 = (const float*)d_in[4];
    float* out = (float*)d_out;

    float* cdf  = (float*)d_ws;           // VOCAB floats
    float* bsum = cdf + VOCAB;            // nb floats
    const int nb = (VOCAB + SCAN_B - 1) / SCAN_B;

    k_partial_sums<<<nb, SCAN_B, 0, stream>>>(unigram75, bsum);
    k_scan_bsums<<<1, 1, 0, stream>>>(bsum, nb);
    k_cdf<<<nb, SCAN_B, 0, stream>>>(unigram75, bsum, cdf);
    k_zero_out<<<1, 1, 0, stream>>>(out);
    k_sgns<<<BATCH / WPB, WPB * 32, 0, stream>>>(centers, pos, embed_in, embed_out,
                                                 cdf, out);
}